// SelfAttentionRouter_53515292508493
// MI455X (gfx1250) — compile-verified
//
#include <hip/hip_runtime.h>

typedef __attribute__((ext_vector_type(2))) float v2f;
typedef __attribute__((ext_vector_type(8))) float v8f;

#define EPSF   1e-6f
#define SCALEF 0.35355339059327373f   // 1/sqrt(8)
#define D_FEAT 2048
#define KB_ITERS 512                  // 2048 / 4

// -------------------------------------------------------------------------
// Prologue: repack W (2048x8 row-major) into WMMA f32 16x16x4 A-fragment
// order. A = W^T tile: lane L (m = L&15, koff = (L>>4)*2) holds
// { W[(4*kb+koff)*8+m], W[(4*kb+koff+1)*8+m] } (zero for m>=8).
// Table layout: T[kb][lane][2] floats -> hot loop reads one coalesced b64.
// -------------------------------------------------------------------------
__global__ void swizzle_w_kernel(const float* __restrict__ Wq,
                                 const float* __restrict__ Wk,
                                 float* __restrict__ Aq,
                                 float* __restrict__ Ak) {
    int idx = blockIdx.x * blockDim.x + threadIdx.x;   // 0..32767
    int e    = idx & 16383;                            // entry within matrix
    int kb   = e >> 5;
    int L    = e & 31;
    int m    = L & 15;
    int kk   = kb * 4 + ((L >> 4) << 1);
    const float* W = (idx < 16384) ? Wq : Wk;
    float*       T = (idx < 16384) ? Aq : Ak;
    float t0 = 0.0f, t1 = 0.0f;
    if (m < 8) {
        t0 = W[kk * 8 + m];
        t1 = W[(kk + 1) * 8 + m];
    }
    T[(size_t)kb * 64 + L * 2 + 0] = t0;
    T[(size_t)kb * 64 + L * 2 + 1] = t1;
}

// -------------------------------------------------------------------------
// Core: one wave processes 16 data rows (stride 2048).
// D = A(W^T) x B(rows^T) accumulated over K=2048 via v_wmma_f32_16x16x4_f32.
// Result: lane n (n<16) holds dot(row n, W[:,h]) in c[h], h=0..7.
// ss accumulates per-lane partial sum of squares (lane L covers k%4 in
// {off,off+1}); merge with shfl_xor(16) outside.
// -------------------------------------------------------------------------
__device__ __forceinline__ void gemv_tile16(const float* __restrict__ rows,
                                            const float* __restrict__ Atab,
                                            int lane, v8f& cOut, float& ssOut) {
    const int m   = lane & 15;
    const int off = (lane >> 4) << 1;
    const float* rp = rows + (size_t)m * D_FEAT + off;
    const float* ap = Atab + lane * 2;
    v8f   acc = {};
    float ss  = 0.0f;
#pragma unroll 4
    for (int kb = 0; kb < KB_ITERS; ++kb) {
        v2f a = *(const v2f*)(ap + (size_t)kb * 64);
        v2f b = *(const v2f*)(rp + (size_t)kb * 4);
        ss = fmaf(b.x, b.x, ss);
        ss = fmaf(b.y, b.y, ss);
        // D = A(16x4) * B(4x16) + C   (f32, full precision)
        acc = __builtin_amdgcn_wmma_f32_16x16x4_f32(
            false, a, false, b, (short)0, acc, false, false);
    }
    cOut  = acc;
    ssOut = ss;
}

// -------------------------------------------------------------------------
// Kernel A: queries = l2norm(x) @ W_q + b_q  -> ws_q [8192 x 8] f32
// One wave per 16 x-rows. 8192 rows -> 512 waves.
// -------------------------------------------------------------------------
__global__ void q_proj_kernel(const float* __restrict__ x,
                              const float* __restrict__ Aq,
                              const float* __restrict__ bq,
                              float* __restrict__ ws_q) {
    int wave    = blockIdx.x * (blockDim.x >> 5) + (threadIdx.x >> 5);
    int lane    = threadIdx.x & 31;
    int rowBase = wave * 16;

    v8f c; float ss;
    gemv_tile16(x + (size_t)rowBase * D_FEAT, Aq, lane, c, ss);

    ss += __shfl_xor(ss, 16);
    float rinv = rsqrtf(ss + EPSF);

    if (lane < 16) {
        float o[8];
#pragma unroll
        for (int h = 0; h < 8; ++h) o[h] = fmaf(c[h], rinv, bq[h]);
        float4* dst = (float4*)(ws_q + (size_t)(rowBase + lane) * 8);
        float4 lo = {o[0], o[1], o[2], o[3]};
        float4 hi = {o[4], o[5], o[6], o[7]};
        dst[0] = lo;
        dst[1] = hi;
    }
}

// -------------------------------------------------------------------------
// Kernel B: keys = l2norm(bax) @ W_k + b_k ; scores = q . k * SCALE ;
// softmax over N=8 -> out [B*S*N].
// One wave per 16 bax-rows (= 2 (b,s) positions). 65536 rows -> 4096 waves.
// -------------------------------------------------------------------------
__global__ void k_attn_kernel(const float* __restrict__ bax,
                              const float* __restrict__ Ak,
                              const float* __restrict__ bk,
                              const float* __restrict__ ws_q,
                              float* __restrict__ out) {
    int wave    = blockIdx.x * (blockDim.x >> 5) + (threadIdx.x >> 5);
    int lane    = threadIdx.x & 31;
    int rowBase = wave * 16;

    v8f c; float ss;
    gemv_tile16(bax + (size_t)rowBase * D_FEAT, Ak, lane, c, ss);

    ss += __shfl_xor(ss, 16);
    float rinv = rsqrtf(ss + EPSF);

    int n   = lane & 15;
    int pos = (rowBase >> 3) + (n >> 3);     // (b*S+s) position index
    float4 q0 = *(const float4*)(ws_q + (size_t)pos * 8);
    float4 q1 = *(const float4*)(ws_q + (size_t)pos * 8 + 4);
    float qv[8] = {q0.x, q0.y, q0.z, q0.w, q1.x, q1.y, q1.z, q1.w};

    float score = 0.0f;
#pragma unroll
    for (int h = 0; h < 8; ++h) {
        float kh = fmaf(c[h], rinv, bk[h]);  // key element
        score = fmaf(qv[h], kh, score);
    }
    score *= SCALEF;

    // softmax across the 8 lanes of each position (xor 1/2/4 stays in-group)
    float mx = score;
    mx = fmaxf(mx, __shfl_xor(mx, 1));
    mx = fmaxf(mx, __shfl_xor(mx, 2));
    mx = fmaxf(mx, __shfl_xor(mx, 4));
    float e = __expf(score - mx);
    float sum = e;
    sum += __shfl_xor(sum, 1);
    sum += __shfl_xor(sum, 2);
    sum += __shfl_xor(sum, 4);

    if (lane < 16) out[rowBase + lane] = e / sum;
}

// -------------------------------------------------------------------------
// Launch. Inputs (setup_inputs order): x, bax, W_q, b_q, W_k, b_k (all f32).
// Workspace (floats): [0,32768) Aq table | [32768,65536) Ak table |
//                     [65536,131072) queries. Total 512 KiB.
// -------------------------------------------------------------------------
extern "C" void kernel_launch(void* const* d_in, const int* in_sizes, int n_in,
                              void* d_out, int out_size, void* d_ws, size_t ws_size,
                              hipStream_t stream) {
    const float* x   = (const float*)d_in[0];
    const float* bax = (const float*)d_in[1];
    const float* Wq  = (const float*)d_in[2];
    const float* bq  = (const float*)d_in[3];
    const float* Wk  = (const float*)d_in[4];
    const float* bk  = (const float*)d_in[5];
    float*       out = (float*)d_out;

    float* ws  = (float*)d_ws;
    float* Aq  = ws;             // 32768 floats
    float* Ak  = ws + 32768;     // 32768 floats
    float* wsq = ws + 65536;     // 65536 floats (queries)

    swizzle_w_kernel<<<128, 256, 0, stream>>>(Wq, Wk, Aq, Ak);
    q_proj_kernel<<<64, 256, 0, stream>>>(x, Aq, bq, wsq);
    k_attn_kernel<<<512, 256, 0, stream>>>(bax, Ak, bk, wsq, out);
}